// SuperVoxelLoss_75136157877152
// MI455X (gfx1250) — compile-verified
//
#include <hip/hip_runtime.h>
#include <math.h>

// Problem geometry (fixed by the reference): B=2, C=2, 128^3 volume.
#define VOL   (1u << 21)      // 128*128*128
#define NVOX  (2u * VOL)      // batch * volume
#define NLBL  (4u * VOL)      // 4 label volumes: pred b0, pred b1, tgt b0, tgt b1
#define NPART 16384u          // NVOX / 256 block partial sums
#define NITER 48

typedef __attribute__((ext_vector_type(2))) float v2f;
typedef __attribute__((ext_vector_type(8))) float v8f;

// ---------------------------------------------------------------------------
// Init: labels = mask ? linear_index+1 : 0  (per volume).
// pred mask = preds[:,1] > 0 ; tgt mask = targets > 0.
// Label volume layout: v=0,1 -> pred batch v ; v=2,3 -> tgt batch v-2.
// ---------------------------------------------------------------------------
__global__ __launch_bounds__(256) void svl_init_labels(
    const float* __restrict__ preds, const int* __restrict__ targets,
    int* __restrict__ lbl) {
  unsigned g = blockIdx.x * 256u + threadIdx.x;          // [0, NVOX)
  unsigned b = g >> 21, i = g & (VOL - 1u);
  float p1 = preds[(b * 2u + 1u) * VOL + i];
  int   t  = targets[g];
  int   id = (int)(i + 1u);
  lbl[b * VOL + i]          = (p1 > 0.0f) ? id : 0;      // pred labels
  lbl[(2u + b) * VOL + i]   = (t  > 0)    ? id : 0;      // tgt labels
}

// ---------------------------------------------------------------------------
// Propagation pass A: m = mask ? max(self, 6 in-bounds neighbors) : 0
// (zero-padded shifts in the reference are a no-op under max of labels >= 0;
//  mask == (lbl > 0) is an invariant of the iteration).
// ---------------------------------------------------------------------------
__global__ __launch_bounds__(256) void svl_prop_max(
    const int* __restrict__ lbl, int* __restrict__ tmp) {
  unsigned t = blockIdx.x * 256u + threadIdx.x;          // [0, NLBL)
  unsigned v = t >> 21, i = t & (VOL - 1u);
  const int* base = lbl + (v << 21);
  int c = base[i];
  int m = 0;
  if (c > 0) {
    m = c;
    unsigned x = i & 127u, y = (i >> 7) & 127u, z = i >> 14;
    if (x > 0u)    m = max(m, base[i - 1u]);
    if (x < 127u)  m = max(m, base[i + 1u]);
    if (y > 0u)    m = max(m, base[i - 128u]);
    if (y < 127u)  m = max(m, base[i + 128u]);
    if (z > 0u)    m = max(m, base[i - 16384u]);
    if (z < 127u)  m = max(m, base[i + 16384u]);
  }
  tmp[t] = m;
}

// ---------------------------------------------------------------------------
// Propagation pass B (pointer jump): out = mask ? max(m, m[m-1]) : 0
// Gather is within the same volume, from the fully-materialized m array.
// ---------------------------------------------------------------------------
__global__ __launch_bounds__(256) void svl_prop_jump(
    const int* __restrict__ tmp, int* __restrict__ lbl) {
  unsigned t = blockIdx.x * 256u + threadIdx.x;          // [0, NLBL)
  unsigned v = t >> 21;
  int m = tmp[t];
  int out = 0;
  if (m > 0) {
    int hop = tmp[(v << 21) + (unsigned)(m - 1)];
    out = max(m, hop);
  }
  lbl[t] = out;
}

// ---------------------------------------------------------------------------
// Zero the flag arrays (reuses the tmp buffer after CCL converges).
// ---------------------------------------------------------------------------
__global__ __launch_bounds__(256) void svl_clear_flags(int* __restrict__ flags) {
  unsigned t = blockIdx.x * 256u + threadIdx.x;
  flags[t] = 0;
}

// ---------------------------------------------------------------------------
// Scatter per-component error flags (segment_max of err == idempotent
// stores of 1; races are benign). flags indexed by (label-1) per volume:
//   m_neg uses tgt labels (volumes 2,3), error where tgt fg & pred bg
//   m_pos uses pred labels (volumes 0,1), error where pred fg & tgt bg
// ---------------------------------------------------------------------------
__global__ __launch_bounds__(256) void svl_scatter_err(
    const int* __restrict__ lbl, int* __restrict__ flags) {
  unsigned g = blockIdx.x * 256u + threadIdx.x;          // [0, NVOX)
  unsigned b = g >> 21, i = g & (VOL - 1u);
  int pl = lbl[b * VOL + i];
  int tl = lbl[(2u + b) * VOL + i];
  if (tl > 0 && pl == 0) flags[((2u + b) << 21) + (unsigned)(tl - 1)] = 1;
  if (pl > 0 && tl == 0) flags[(b << 21) + (unsigned)(pl - 1)] = 1;
}

// ---------------------------------------------------------------------------
// Stage 1: per-voxel loss * blend factor, block-reduced to one partial sum.
// ce = softplus(p_other - p_tgt); factor = 0.5 + 0.25*m_neg + 0.25*m_pos
// (ALPHA = BETA = 0.5 folded in). Deterministic wave32 + LDS tree reduction.
// ---------------------------------------------------------------------------
__global__ __launch_bounds__(256) void svl_stage1(
    const float* __restrict__ preds, const int* __restrict__ targets,
    const int* __restrict__ lbl, const int* __restrict__ flags,
    float* __restrict__ partials) {
  unsigned g = blockIdx.x * 256u + threadIdx.x;          // [0, NVOX)
  unsigned b = g >> 21, i = g & (VOL - 1u);

  float p0 = preds[(b * 2u) * VOL + i];
  float p1 = preds[(b * 2u + 1u) * VOL + i];
  int   t  = targets[g];
  float pt = (t > 0) ? p1 : p0;
  float po = (t > 0) ? p0 : p1;
  float d  = po - pt;
  float ce = fmaxf(d, 0.0f) + log1pf(expf(-fabsf(d)));   // -log_softmax[t]

  int pl = lbl[b * VOL + i];
  int tl = lbl[(2u + b) * VOL + i];
  float fneg = (tl > 0 && flags[((2u + b) << 21) + (unsigned)(tl - 1)]) ? 1.0f : 0.0f;
  float fpos = (pl > 0 && flags[(b << 21) + (unsigned)(pl - 1)]) ? 1.0f : 0.0f;
  float val = ce * (0.5f + 0.25f * fneg + 0.25f * fpos);

  // wave32 reduction
  for (int off = 16; off > 0; off >>= 1)
    val += __shfl_xor(val, off, 32);

  __shared__ float smem[8];
  unsigned lane = threadIdx.x & 31u, wid = threadIdx.x >> 5;
  if (lane == 0u) smem[wid] = val;
  __syncthreads();
  if (threadIdx.x == 0u) {
    float acc = 0.0f;
    for (int w = 0; w < 8; ++w) acc += smem[w];          // fixed order: deterministic
    partials[blockIdx.x] = acc;
  }
}

// ---------------------------------------------------------------------------
// Stage 2: single-wave WMMA reduction of the 16384 partials.
// A = ones(16x4) so D[m][n] = sum_k B[k][n] for every m; any placement of the
// 64 values into B's 64 lane-slots yields the same total. Summing acc[0]
// across all 32 lanes gives 2x the grand total (M=0 and M=8 rows identical).
// EXEC is all 1s: exactly one full wave, no divergence.
// ---------------------------------------------------------------------------
__global__ __launch_bounds__(32) void svl_stage2(
    const float* __restrict__ partials, float* __restrict__ out) {
  unsigned lane = threadIdx.x;                            // 0..31
  v2f a; a.x = 1.0f; a.y = 1.0f;
  v8f acc = {};
  for (unsigned it = 0; it < NPART / 64u; ++it) {
    v2f bv;
    bv.x = partials[it * 64u + lane * 2u + 0u];
    bv.y = partials[it * 64u + lane * 2u + 1u];
    // D = A(ones) x B + C   ->  column-sum accumulator
    acc = __builtin_amdgcn_wmma_f32_16x16x4_f32(
        /*neg_a=*/false, a, /*neg_b=*/false, bv,
        /*c_mod=*/(short)0, acc, /*reuse_a=*/false, /*reuse_b=*/false);
  }
  float v = acc[0];
  for (int off = 16; off > 0; off >>= 1)
    v += __shfl_xor(v, off, 32);                          // = 2 * total
  if (lane == 0u)
    out[0] = v * 0.5f * (1.0f / (float)NVOX);             // mean
}

// ---------------------------------------------------------------------------
extern "C" void kernel_launch(void* const* d_in, const int* in_sizes, int n_in,
                              void* d_out, int out_size, void* d_ws, size_t ws_size,
                              hipStream_t stream) {
  const float* preds   = (const float*)d_in[0];  // (2,2,128,128,128) f32
  const int*   targets = (const int*)d_in[1];    // (2,1,128,128,128) i32

  char* ws = (char*)d_ws;
  int*   lbl      = (int*)ws;                                 // 4*VOL i32 (32 MB)
  int*   tmp      = (int*)(ws + (size_t)NLBL * sizeof(int));  // 4*VOL i32 (32 MB), reused as flags
  float* partials = (float*)(ws + (size_t)2 * NLBL * sizeof(int)); // 16384 f32

  const dim3 blk(256);
  const unsigned gVox = NVOX / 256u;   // 16384
  const unsigned gLbl = NLBL / 256u;   // 32768

  svl_init_labels<<<gVox, blk, 0, stream>>>(preds, targets, lbl);

  for (int it = 0; it < NITER; ++it) {
    svl_prop_max <<<gLbl, blk, 0, stream>>>(lbl, tmp);
    svl_prop_jump<<<gLbl, blk, 0, stream>>>(tmp, lbl);
  }

  svl_clear_flags<<<gLbl, blk, 0, stream>>>(tmp);
  svl_scatter_err<<<gVox, blk, 0, stream>>>(lbl, tmp);
  svl_stage1<<<gVox, blk, 0, stream>>>(preds, targets, lbl, tmp, partials);
  svl_stage2<<<1, 32, 0, stream>>>(partials, (float*)d_out);
}